// amp_net_unentangled_89146341196029
// MI455X (gfx1250) — compile-verified
//
#include <hip/hip_runtime.h>
#include <stdint.h>

// amp_net_unentangled: out[b,k,c] = prod_{i=0..11} amp[c*12+i, (k>>(11-i))&1]
// Batch dimension is a pure broadcast -> compute each 64x64 k-tile once in LDS,
// then DMA-replicate it to the 256 batch slots with the CDNA5 Tensor Data Mover.
// Pure store-bandwidth problem: 256 MB out / 23.3 TB/s ~= 11 us floor.

typedef __attribute__((ext_vector_type(4))) unsigned int v4u;
typedef __attribute__((ext_vector_type(8))) int          v8i;
typedef __attribute__((ext_vector_type(4))) int          v4i;

#define CELLSIZE 12
#define NCELL    64
#define KTOT     4096            // 2^CELLSIZE
#define NBATCH   256
#define KTILE    64              // k-rows per workgroup tile (16 KB tile)
#define BSPLIT   4               // batch groups (parallelism across batches)
#define BPG      (NBATCH / BSPLIT)   // 64 batches per workgroup
#define TPB      256
#define NWAVES   (TPB / 32)          // 8 waves
#define BPW      (BPG / NWAVES)      // 8 batches per wave (8 TDM ops/wave)

__global__ __launch_bounds__(TPB) void amp_kron_tdm_bcast(
    const float* __restrict__ amp, float* __restrict__ out)
{
    __shared__ float s_amp[CELLSIZE * NCELL * 2];   // 6 KB: amp staged in LDS
    __shared__ float s_tile[KTILE * NCELL];         // 16 KB: one k-slice of T

    const int tid = (int)threadIdx.x;
    const int kg  = (int)(blockIdx.x & (KTOT / KTILE - 1));  // 64 k-groups
    const int bg  = (int)(blockIdx.x >> 6);                  // 4 batch-groups
    const int k0  = kg * KTILE;
    const int b0  = bg * BPG;

    // ---- Phase 0: stage amp (768x2 f32) into LDS ----
    for (int i = tid; i < CELLSIZE * NCELL * 2; i += TPB) s_amp[i] = amp[i];
    __syncthreads();

    // ---- Phase 1: build T[k0:k0+64, 0:64] in LDS (16 elements / thread) ----
    #pragma unroll
    for (int j = 0; j < (KTILE * NCELL) / TPB; ++j) {
        const int idx = j * TPB + tid;          // [k_local*64 + c]
        const int c   = idx & (NCELL - 1);
        const int k   = k0 + (idx >> 6);
        const float* row = &s_amp[c * CELLSIZE * 2];
        float p = row[(k >> 11) & 1];           // i = 0 (MSB of k)
        #pragma unroll
        for (int i = 1; i < CELLSIZE; ++i)
            p *= row[i * 2 + ((k >> (11 - i)) & 1)];
        s_tile[idx] = p;
    }
    __syncthreads();

    // ---- Phase 2: TDM broadcast, distributed over all 8 waves.
    // Each wave DMAs the shared 16 KB tile to its own 8 batch slots (disjoint
    // destinations, so cross-wave TDM unordering is harmless). All phase-2
    // values are wave-uniform (readfirstlane-forced scalar), so each wave's
    // instruction stream issues exactly its own descriptors (TDM ignores EXEC).
    {
        const int w = __builtin_amdgcn_readfirstlane(tid) >> 5;  // wave id 0..7

        // LDS aperture: generic addr = {SHARED_BASE, offset32} -> low 32 bits
        const uint32_t lds_off = (uint32_t)(uintptr_t)(const void*)s_tile;
        const uint64_t gbase = (uint64_t)(uintptr_t)out
            + ((uint64_t)(b0 + w * BPW) * KTOT + (uint64_t)k0)
              * NCELL * sizeof(float);

        const unsigned ELEMS = KTILE * NCELL;   // 4096 elems = 16 KB, contiguous

        // D# group 1 (ISA 8.4): wg_mask=0 | data_size=2(4B);
        // tensor_dim0 = tile_dim0 = 4096 (1-D tile); tensor_dim1 = 1;
        // tile_dim1/2 = 0 (unused); dim0 stride = 4096.
        v8i g1;
        g1[0] = (int)(2u << 16);                       // data_size = 4 bytes
        g1[1] = (int)((ELEMS & 0xFFFFu) << 16);        // tensor_dim0[15:0]
        g1[2] = (int)(1u << 16);                       // tensor_dim0 hi=0 | tensor_dim1=1
        g1[3] = (int)((ELEMS & 0xFFFFu) << 16);        // tensor_dim1 hi=0 | tile_dim0
        g1[4] = 0;                                     // tile_dim1=0, tile_dim2=0
        g1[5] = (int)ELEMS;                            // tensor_dim0_stride[31:0]
        g1[6] = 0;                                     // stride0 hi | stride1 lo
        g1[7] = 0;                                     // stride1 hi
        // D# groups 2/3 (8.5/8.6): higher dims unused (tile_dim1==0)
        v4i g2; g2[0] = 1; g2[1] = 1; g2[2] = 0; g2[3] = 0;
        v4i g3; g3[0] = 0; g3[1] = 0; g3[2] = 0; g3[3] = 0;
        // 6-arg toolchain variant: 5th arg is an (uncharacterized) int32x8
        // extension group -- zero-filled (probe-verified to compile as zeros).
        v8i g4; g4[0] = 0; g4[1] = 0; g4[2] = 0; g4[3] = 0;
                g4[4] = 0; g4[5] = 0; g4[6] = 0; g4[7] = 0;

        #pragma unroll
        for (int b = 0; b < BPW; ++b) {
            // Per batch, out[b, k0:k0+64, :] is one contiguous 16 KB block.
            const uint64_t ga = gbase
                + (uint64_t)b * (uint64_t)(KTOT * NCELL * sizeof(float)); // +1 MB
            // D# group 0 (ISA 8.3): count=1 | lds_addr | global_addr | type=2
            v4u g0;
            g0[0] = 1u;                                          // count=1, no gather
            g0[1] = lds_off;                                     // lds_addr (bytes)
            g0[2] = (uint32_t)ga;                                // global_addr[31:0]
            g0[3] = (uint32_t)((ga >> 32) & 0x1FFFFFFu)          // global_addr[56:32]
                  | (2u << 30);                                  // type = 2 ("image")
            __builtin_amdgcn_tensor_store_from_lds(g0, g1, g2, g3, g4, /*cpol=*/0);
        }
        __builtin_amdgcn_s_wait_tensorcnt(0);  // LDS must stay live until DMA done
    }
}

extern "C" void kernel_launch(void* const* d_in, const int* in_sizes, int n_in,
                              void* d_out, int out_size, void* d_ws, size_t ws_size,
                              hipStream_t stream) {
    (void)in_sizes; (void)n_in; (void)d_ws; (void)ws_size; (void)out_size;
    // d_in[0] = inputs (values unused by the reference), d_in[1] = amp (768x2 f32)
    const float* amp = (const float*)d_in[1];
    float*       out = (float*)d_out;
    amp_kron_tdm_bcast<<<dim3((KTOT / KTILE) * BSPLIT), dim3(TPB), 0, stream>>>(amp, out);
}